// PseudoInstanceDiscriminativeEnhancement_17935783428230
// MI455X (gfx1250) — compile-verified
//
#include <hip/hip_runtime.h>
#include <math.h>

typedef float v2f __attribute__((ext_vector_type(2)));
typedef float v8f __attribute__((ext_vector_type(8)));

#define B_ 32
#define T_ 1024
#define D_ 512
#define INV_TEMP 10.0f      // 1 / 0.1
#define EPS_NORM 1e-12f
#define EPS_LOG  1e-8f

// ---------------- Kernel A: per-batch argmax/argmin + query squared norms ----
__global__ void __launch_bounds__(256)
pide_prep_kernel(const float* __restrict__ scores,
                 const int*   __restrict__ seq_len,
                 const float* __restrict__ features,
                 int*   __restrict__ ws_idx,
                 float* __restrict__ ws_qn) {
  const int b   = blockIdx.x;
  const int tid = threadIdx.x;
  const int L   = seq_len[b];
  const float* sc = scores + (size_t)b * T_;

  float bmax = -__builtin_inff(); int imax = 0;
  float bmin =  __builtin_inff(); int imin = 0;
  for (int t = tid; t < T_; t += 256) {
    float v  = sc[t];
    float vx = (t < L) ? v : -__builtin_inff();
    float vn = (t < L) ? v :  __builtin_inff();
    if (vx > bmax) { bmax = vx; imax = t; }   // strict -> first occurrence
    if (vn < bmin) { bmin = vn; imin = t; }
  }

  __shared__ float sv[256];
  __shared__ int   si[256];

  // argmax reduce (ties -> smaller index, matching jnp.argmax)
  sv[tid] = bmax; si[tid] = imax; __syncthreads();
  for (int s = 128; s > 0; s >>= 1) {
    if (tid < s) {
      float v2 = sv[tid + s]; int i2 = si[tid + s];
      if (v2 > sv[tid] || (v2 == sv[tid] && i2 < si[tid])) { sv[tid] = v2; si[tid] = i2; }
    }
    __syncthreads();
  }
  const int idx1 = si[0];
  __syncthreads();

  // argmin reduce
  sv[tid] = bmin; si[tid] = imin; __syncthreads();
  for (int s = 128; s > 0; s >>= 1) {
    if (tid < s) {
      float v2 = sv[tid + s]; int i2 = si[tid + s];
      if (v2 < sv[tid] || (v2 == sv[tid] && i2 < si[tid])) { sv[tid] = v2; si[tid] = i2; }
    }
    __syncthreads();
  }
  const int idx2 = si[0];
  __syncthreads();

  // squared norms of the two query rows
  const float* q1 = features + ((size_t)b * T_ + idx1) * D_;
  const float* q2 = features + ((size_t)b * T_ + idx2) * D_;
  float p1 = 0.f, p2 = 0.f;
  for (int d = tid; d < D_; d += 256) {
    float x = q1[d]; p1 += x * x;
    float y = q2[d]; p2 += y * y;
  }
  sv[tid] = p1; __syncthreads();
  for (int s = 128; s > 0; s >>= 1) { if (tid < s) sv[tid] += sv[tid + s]; __syncthreads(); }
  const float qn1 = sv[0]; __syncthreads();
  sv[tid] = p2; __syncthreads();
  for (int s = 128; s > 0; s >>= 1) { if (tid < s) sv[tid] += sv[tid + s]; __syncthreads(); }
  const float qn2 = sv[0];

  if (tid == 0) {
    ws_idx[b * 2 + 0] = idx1;
    ws_idx[b * 2 + 1] = idx2;
    ws_qn [b * 2 + 0] = qn1;
    ws_qn [b * 2 + 1] = qn2;
  }
}

// ------- Kernel B: streaming WMMA pass, partial exp-sums per (batch, chunk) --
__global__ void __launch_bounds__(256)
pide_sim_kernel(const float* __restrict__ features,
                const int*   __restrict__ ws_idx,
                const float* __restrict__ ws_qn,
                float* __restrict__ ws_S) {
  const int b     = blockIdx.y;
  const int chunk = blockIdx.x;          // 8 chunks of 128 rows
  const int wave  = threadIdx.x >> 5;    // 8 waves, one 16-row tile each
  const int lane  = threadIdx.x & 31;
  const int m     = lane & 15;           // A-matrix row within tile / B column
  const int kh    = lane >> 4;           // which K half-pair this lane carries
  const int row   = chunk * 128 + wave * 16 + m;

  const int idx1 = ws_idx[b * 2 + 0];
  const int idx2 = ws_idx[b * 2 + 1];

  const float* base = features + (size_t)b * T_ * D_;
  const float* rowp = base + (size_t)row * D_ + kh * 2;
  const float* qp   = base + (size_t)(m == 0 ? idx1 : idx2) * D_ + kh * 2;
  const float maskf = (m < 2) ? 1.0f : 0.0f;   // B columns >= 2 are zero

  v8f cC = {};   // rows . [q1 q2 0 ...]
  v8f cG = {};   // Gram(tile): diag = row squared norms (B layout == A layout!)
#pragma unroll 8
  for (int k = 0; k < D_; k += 4) {
    v2f a = *(const v2f*)(rowp + k);     // A[m][kh*2 + {0,1}]
    v2f q = *(const v2f*)(qp   + k);     // B[kh*2 + {0,1}][m]
    q.x *= maskf; q.y *= maskf;
    cC = __builtin_amdgcn_wmma_f32_16x16x4_f32(false, a, false, q, (short)0, cC, false, false);
    cG = __builtin_amdgcn_wmma_f32_16x16x4_f32(false, a, false, a, (short)0, cG, false, false);
  }

  // D layout: D[m][n] -> VGPR (m%8), lane n + 16*(m>=8)
  __shared__ float ldsC[8][32][8];
  __shared__ float ldsG[8][32][8];
  __shared__ float ldsS[8][2];
  *(v8f*)(&ldsC[wave][lane][0]) = cC;
  *(v8f*)(&ldsG[wave][lane][0]) = cG;
  __syncthreads();

  float e1 = 0.f, e2 = 0.f;
  if (lane < 16) {
    const int i = lane;                        // tile row
    float c1, c2, g;
    if (i < 8) {
      c1 = ldsC[wave][0][i];                   // D[i][0]
      c2 = ldsC[wave][1][i];                   // D[i][1]
      g  = ldsG[wave][i][i];                   // G[i][i]
    } else {
      c1 = ldsC[wave][16][i - 8];
      c2 = ldsC[wave][17][i - 8];
      g  = ldsG[wave][i + 16][i - 8];
    }
    const float dns = fmaxf(sqrtf(g), EPS_NORM);
    const float dn1 = fmaxf(sqrtf(ws_qn[b * 2 + 0]), EPS_NORM);
    const float dn2 = fmaxf(sqrtf(ws_qn[b * 2 + 1]), EPS_NORM);
    e1 = expf(c1 / (dn1 * dns) * INV_TEMP);
    e2 = expf(c2 / (dn2 * dns) * INV_TEMP);
  }
  for (int off = 16; off > 0; off >>= 1) {
    e1 += __shfl_xor(e1, off, 32);
    e2 += __shfl_xor(e2, off, 32);
  }
  if (lane == 0) { ldsS[wave][0] = e1; ldsS[wave][1] = e2; }
  __syncthreads();
  if (threadIdx.x == 0) {
    float S1 = 0.f, S2 = 0.f;
    for (int w = 0; w < 8; ++w) { S1 += ldsS[w][0]; S2 += ldsS[w][1]; }
    ws_S[((size_t)b * 8 + chunk) * 2 + 0] = S1;   // deterministic slot, no atomics
    ws_S[((size_t)b * 8 + chunk) * 2 + 1] = S2;
  }
}

// ---------------- Kernel C: finalize loss ------------------------------------
__global__ void pide_final_kernel(const int*   __restrict__ ws_idx,
                                  const float* __restrict__ ws_qn,
                                  const float* __restrict__ ws_S,
                                  float* __restrict__ out) {
  const int b = threadIdx.x;   // 32 lanes, one per batch
  float num = 0.f, cnt = 0.f;
  {
    const int   i1  = ws_idx[b * 2 + 0];
    const int   i2  = ws_idx[b * 2 + 1];
    const float qn1 = ws_qn[b * 2 + 0];
    const float qn2 = ws_qn[b * 2 + 1];
    float S1 = 0.f, S2 = 0.f;
    for (int c = 0; c < 8; ++c) {
      S1 += ws_S[((size_t)b * 8 + c) * 2 + 0];
      S2 += ws_S[((size_t)b * 8 + c) * 2 + 1];
    }
    const float dn1 = fmaxf(sqrtf(qn1), EPS_NORM);
    const float dn2 = fmaxf(sqrtf(qn2), EPS_NORM);
    const float lp1 = (qn1 / (dn1 * dn1)) * INV_TEMP - logf(S1 + EPS_LOG);
    const float lp2 = (qn2 / (dn2 * dn2)) * INV_TEMP - logf(S2 + EPS_LOG);
    if (i1 == i2) { num = lp1;       cnt = 1.f; }   // -1 overwrote +1: one pos pair
    else          { num = lp1 + lp2; cnt = 2.f; }
  }
  for (int off = 16; off > 0; off >>= 1) {
    num += __shfl_xor(num, off, 32);
    cnt += __shfl_xor(cnt, off, 32);
  }
  if (b == 0) out[0] = -num / (cnt + EPS_LOG);
}

extern "C" void kernel_launch(void* const* d_in, const int* in_sizes, int n_in,
                              void* d_out, int out_size, void* d_ws, size_t ws_size,
                              hipStream_t stream) {
  (void)in_sizes; (void)n_in; (void)out_size; (void)ws_size;
  const float* features = (const float*)d_in[0];   // [32,1024,512] f32
  const float* scores   = (const float*)d_in[1];   // [32,1024,1]  f32
  const int*   seq_len  = (const int*)  d_in[2];   // [32]         i32
  float* out = (float*)d_out;

  char* ws = (char*)d_ws;
  int*   ws_idx = (int*)(ws + 0);       // 64 ints
  float* ws_qn  = (float*)(ws + 256);   // 64 floats
  float* ws_S   = (float*)(ws + 512);   // 32*8*2 floats (rewritten every call)

  pide_prep_kernel <<<B_, 256, 0, stream>>>(scores, seq_len, features, ws_idx, ws_qn);
  pide_sim_kernel  <<<dim3(8, B_), 256, 0, stream>>>(features, ws_idx, ws_qn, ws_S);
  pide_final_kernel<<<1, 32, 0, stream>>>(ws_idx, ws_qn, ws_S, out);
}